// MessageBlock_33225867002060
// MI455X (gfx1250) — compile-verified
//
#include <hip/hip_runtime.h>
#include <math.h>

#define NN 512
#define FF 128
#define F3 384
#define RBFN 20
#define CUTOFF_F 5.0f
#define PI_F 3.14159265358979323846f

typedef __attribute__((ext_vector_type(2))) float v2f;
typedef __attribute__((ext_vector_type(8))) float v8f;

struct f3v { float x, y, z; };

__device__ __forceinline__ v8f wmma_f32_k4(v2f a, v2f b, v8f c) {
    // V_WMMA_F32_16X16X4_F32 : D(16x16,f32) = A(16x4,f32) x B(4x16,f32) + C
    return __builtin_amdgcn_wmma_f32_16x16x4_f32(
        false, a, false, b, (short)0, c, false, false);
}

// ---------------------------------------------------------------------------
// Kernel 1: h = silu(s @ phi_w1 + b1)     s:(512,128) w1:(128,128) h:(512,128)
// ---------------------------------------------------------------------------
__global__ __launch_bounds__(256) void k_mlp1(const float* __restrict__ s,
                                              const float* __restrict__ w1,
                                              const float* __restrict__ b1,
                                              float* __restrict__ h) {
    const int tid   = threadIdx.x;
    const int wave  = tid >> 5;
    const int lane  = tid & 31;
    const int m     = lane & 15;
    const int khalf = (lane < 16) ? 0 : 2;
    const int roff  = (lane < 16) ? 0 : 8;
    const int rbase = blockIdx.x * 16;
    const int cbase = wave * 16;

    v8f acc = {0.f, 0.f, 0.f, 0.f, 0.f, 0.f, 0.f, 0.f};
    const int row = rbase + m;
    for (int k0 = 0; k0 < FF; k0 += 4) {
        v2f a = *(const v2f*)(s + row * FF + k0 + khalf);
        v2f b;
        b.x = w1[(k0 + khalf) * FF + cbase + m];
        b.y = w1[(k0 + khalf + 1) * FF + cbase + m];
        acc = wmma_f32_k4(a, b, acc);
    }
    const float bias = b1[cbase + m];
    for (int r = 0; r < 8; ++r) {
        float x = acc[r] + bias;
        h[(rbase + r + roff) * FF + cbase + m] = x / (1.f + expf(-x));
    }
}

// ---------------------------------------------------------------------------
// Kernel 2: phi = (h @ phi_w2 + b2) * atom_mask[row]
// ---------------------------------------------------------------------------
__global__ __launch_bounds__(256) void k_mlp2(const float* __restrict__ h,
                                              const float* __restrict__ w2,
                                              const float* __restrict__ b2,
                                              const float* __restrict__ mask,
                                              float* __restrict__ phi) {
    const int tid   = threadIdx.x;
    const int wave  = tid >> 5;
    const int lane  = tid & 31;
    const int m     = lane & 15;
    const int khalf = (lane < 16) ? 0 : 2;
    const int roff  = (lane < 16) ? 0 : 8;
    const int rt    = blockIdx.x / 3;
    const int ct    = (blockIdx.x % 3) * 8 + wave;
    const int rbase = rt * 16;
    const int cbase = ct * 16;

    v8f acc = {0.f, 0.f, 0.f, 0.f, 0.f, 0.f, 0.f, 0.f};
    const int row = rbase + m;
    for (int k0 = 0; k0 < FF; k0 += 4) {
        v2f a = *(const v2f*)(h + row * FF + k0 + khalf);
        v2f b;
        b.x = w2[(k0 + khalf) * F3 + cbase + m];
        b.y = w2[(k0 + khalf + 1) * F3 + cbase + m];
        acc = wmma_f32_k4(a, b, acc);
    }
    const float bias = b2[cbase + m];
    for (int r = 0; r < 8; ++r) {
        const int orow = rbase + r + roff;
        phi[orow * F3 + cbase + m] = (acc[r] + bias) * mask[orow];
    }
}

// ---------------------------------------------------------------------------
// Kernel 3: fused message pass. One workgroup per i (512 blocks, 8 waves).
//  - lin_w B-fragments hoisted into 15 v2f registers (loop-invariant).
//  - sin-feature build spread over 160 threads (2 sinf each) and
//    double-buffered in LDS: chunk n+1 features overlap chunk n compute,
//    one barrier per chunk.
// ---------------------------------------------------------------------------
__global__ __launch_bounds__(256) void k_message(const float* __restrict__ v,
                                                 const float* __restrict__ dr,
                                                 const float* __restrict__ pref,
                                                 const float* __restrict__ lin_w,
                                                 const float* __restrict__ lin_b,
                                                 const float* __restrict__ phi,
                                                 float* __restrict__ out_ds,
                                                 float* __restrict__ out_dv) {
    __shared__ float sf_lds[2][16][RBFN]; // sin features (row stride 80B, 8B-aligned)
    __shared__ float e_lds[2][16][3];     // unit vectors e_ij
    __shared__ float g_lds[2][16];        // cosine cutoff
    __shared__ float pf_lds[RBFN];

    const int tid = threadIdx.x;
    const int i   = blockIdx.x;
    if (tid < RBFN) pf_lds[tid] = pref[tid];

    const int  wave  = tid >> 5;
    const int  lane  = tid & 31;
    const int  m     = lane & 15;
    const bool lo    = (lane < 16);
    const int  khalf = lo ? 0 : 2;
    const int  joff  = lo ? 0 : 8;
    const int  c0    = wave * 16;        // msg_ss columns
    const int  c1    = FF + c0;          // msg_vv columns
    const int  c2    = 2 * FF + c0;      // msg_vs columns
    const int  f     = c0 + m;           // feature index this lane owns
    const float lb0 = lin_b[c0 + m];
    const float lb1 = lin_b[c1 + m];
    const float lb2 = lin_b[c2 + m];

    // Hoist all lin_w B-fragments (loop-invariant over j): 5 k-steps x 3 tiles.
    v2f bw0[5], bw1[5], bw2[5];
    for (int kk = 0; kk < 5; ++kk) {
        const int k = 4 * kk + khalf;
        bw0[kk].x = lin_w[k * F3 + c0 + m];
        bw0[kk].y = lin_w[(k + 1) * F3 + c0 + m];
        bw1[kk].x = lin_w[k * F3 + c1 + m];
        bw1[kk].y = lin_w[(k + 1) * F3 + c1 + m];
        bw2[kk].x = lin_w[k * F3 + c2 + m];
        bw2[kk].y = lin_w[(k + 1) * F3 + c2 + m];
    }

    // Fills feature buffer `buf` for chunk starting at `jbase`.
    // 160 threads: jc = tid&15, kx = tid>>4 (0..9) -> 2 sins each.
    auto fill = [&](int buf, int jbase) {
        if (tid < 160) {
            const int jc = tid & 15;
            const int kx = tid >> 4;
            const int j  = jbase + jc;
            const float x = dr[(i * NN + j) * 3 + 0];
            const float y = dr[(i * NN + j) * 3 + 1];
            const float z = dr[(i * NN + j) * 3 + 2];
            const float d = sqrtf(x * x + y * y + z * z);
            sf_lds[buf][jc][kx]      = sinf(d * pf_lds[kx]);
            sf_lds[buf][jc][kx + 10] = sinf(d * pf_lds[kx + 10]);
            if (tid < 16) {
                const float inv = 1.f / (d + 1e-9f);
                e_lds[buf][jc][0] = x * inv;
                e_lds[buf][jc][1] = y * inv;
                e_lds[buf][jc][2] = z * inv;
                g_lds[buf][jc] = (d < CUTOFF_F)
                                   ? 0.5f * (cosf(PI_F * d / CUTOFF_F) + 1.f)
                                   : 0.f;
            }
        }
    };

    float ds_acc = 0.f;
    float dv0 = 0.f, dv1 = 0.f, dv2 = 0.f;

    __syncthreads();        // pf_lds ready
    fill(0, 0);             // prologue: stage chunk 0
    __syncthreads();

    int p = 0;
    for (int jb = 0; jb < NN; jb += 16, p ^= 1) {
        // Stage next chunk into the other buffer (overlaps with compute below).
        // Safe: buf p^1 was last READ one iteration ago, separated by a barrier.
        if (jb + 16 < NN) {
            fill(p ^ 1, jb + 16);
            const int jn = jb + 16 + joff;
            __builtin_prefetch((const void*)(phi + jn * F3 + c0 + m), 0, 1);
            __builtin_prefetch((const void*)(v + (jn * FF + f) * 3), 0, 1);
        }

        v8f a0 = {0.f, 0.f, 0.f, 0.f, 0.f, 0.f, 0.f, 0.f};
        v8f a1 = a0, a2 = a0;
#pragma unroll
        for (int kk = 0; kk < 5; ++kk) {       // K = 20 in steps of 4
            v2f a = *(const v2f*)&sf_lds[p][m][4 * kk + khalf];  // ds_load_b64
            a0 = wmma_f32_k4(a, bw0[kk], a0);
            a1 = wmma_f32_k4(a, bw1[kk], a1);
            a2 = wmma_f32_k4(a, bw2[kk], a2);
        }

        // C layout: VGPR r holds rows M=r (lanes 0-15) / M=r+8 (lanes 16-31)
#pragma unroll
        for (int r = 0; r < 8; ++r) {
            const int jj = r + joff;
            const int j  = jb + jj;
            const float gj = g_lds[p][jj];
            const float* ph = phi + j * F3;
            const float mss = (a0[r] + lb0) * gj * ph[c0 + m];
            const float mvv = (a1[r] + lb1) * gj * ph[c1 + m];
            const float mvs = (a2[r] + lb2) * gj * ph[c2 + m];
            ds_acc += mss;
            const f3v vp = *(const f3v*)(v + (j * FF + f) * 3);
            dv0 += mvv * vp.x + mvs * e_lds[p][jj][0];
            dv1 += mvv * vp.y + mvs * e_lds[p][jj][1];
            dv2 += mvv * vp.z + mvs * e_lds[p][jj][2];
        }
        __syncthreads();   // publish next buffer / retire reads of current
    }

    // merge the two j-halves (lanes l and l+16 hold disjoint j sets)
    ds_acc += __shfl_xor(ds_acc, 16);
    dv0    += __shfl_xor(dv0, 16);
    dv1    += __shfl_xor(dv1, 16);
    dv2    += __shfl_xor(dv2, 16);
    if (lo) {
        out_ds[i * FF + f] = ds_acc;
        float* o = out_dv + (i * FF + f) * 3;
        o[0] = dv0;
        o[1] = dv1;
        o[2] = dv2;
    }
}

extern "C" void kernel_launch(void* const* d_in, const int* in_sizes, int n_in,
                              void* d_out, int out_size, void* d_ws, size_t ws_size,
                              hipStream_t stream) {
    const float* s    = (const float*)d_in[0];
    const float* v    = (const float*)d_in[1];
    const float* dr   = (const float*)d_in[2];
    const float* mask = (const float*)d_in[3];
    const float* pref = (const float*)d_in[4];
    const float* lw   = (const float*)d_in[5];
    const float* lb   = (const float*)d_in[6];
    const float* w1   = (const float*)d_in[7];
    const float* b1   = (const float*)d_in[8];
    const float* w2   = (const float*)d_in[9];
    const float* b2   = (const float*)d_in[10];

    float* ds = (float*)d_out;            // (512,128)
    float* dv = ds + NN * FF;             // (512,128,3)

    float* h   = (float*)d_ws;            // (512,128)  256 KB
    float* phi = h + NN * FF;             // (512,384)  768 KB

    k_mlp1<<<NN / 16, 256, 0, stream>>>(s, w1, b1, h);
    k_mlp2<<<(NN / 16) * 3, 256, 0, stream>>>(h, w2, b2, mask, phi);
    k_message<<<NN, 256, 0, stream>>>(v, dr, pref, lw, lb, phi, ds, dv);
}